// SHEquiFFN_74191265071617
// MI455X (gfx1250) — compile-verified
//
#include <hip/hip_runtime.h>

typedef float v2f __attribute__((ext_vector_type(2)));
typedef float v8f __attribute__((ext_vector_type(8)));

#define TILE   16
#define NTHR   128
#define DIM_IR 288
#define YSTR   292      // padded row stride for sXZ / sY
#define RSTR   1028     // padded row stride for sR

// ================= CG setup (device-side e3nn wigner_3j) =================
struct cplx { double re, im; };
__device__ inline cplx cmul(cplx a, cplx b) {
  return { a.re*b.re - a.im*b.im, a.re*b.im + a.im*b.re };
}
__device__ double dfact(int n) { double r = 1.0; for (int i = 2; i <= n; i++) r *= (double)i; return r; }

__device__ void su2_cg(int j1, int j2, int j3, double* M) {
  int d1 = 2*j1+1, d2 = 2*j2+1, d3 = 2*j3+1;
  for (int a = 0; a < d1*d2*d3; a++) M[a] = 0.0;
  for (int m1 = -j1; m1 <= j1; m1++)
    for (int m2 = -j2; m2 <= j2; m2++) {
      int m3 = m1 + m2;
      if (m3 < -j3 || m3 > j3) continue;
      double pref = sqrt((double)(2*j3+1) * dfact(j3+j1-j2) * dfact(j3-j1+j2)
                         * dfact(j1+j2-j3) / dfact(j1+j2+j3+1));
      pref *= sqrt(dfact(j3+m3)*dfact(j3-m3)*dfact(j1-m1)*dfact(j1+m1)
                   *dfact(j2-m2)*dfact(j2+m2));
      double s = 0.0;
      for (int k = 0; k <= j1+j2-j3; k++) {
        int dd[6] = { k, j1+j2-j3-k, j1-m1-k, j2+m2-k, j3-j2+m1+k, j3-j1-m2+k };
        bool bad = false; double pr = 1.0;
        for (int q = 0; q < 6; q++) { if (dd[q] < 0) { bad = true; break; } pr *= dfact(dd[q]); }
        if (bad) continue;
        s += ((k & 1) ? -1.0 : 1.0) / pr;
      }
      M[((m1+j1)*d2 + (m2+j2))*d3 + (m3+j3)] = pref * s;
    }
}

__device__ void qmat(int l, cplx* q) {
  int d = 2*l+1;
  for (int a = 0; a < d*d; a++) q[a] = {0.0, 0.0};
  const double is2 = 0.70710678118654752440;
  for (int m = -l; m < 0; m++) {
    int am = -m;
    q[(l+m)*d + (l+am)] = { is2, 0.0 };
    q[(l+m)*d + (l-am)] = { 0.0, -is2 };
  }
  q[l*d + l] = { 1.0, 0.0 };
  for (int m = 1; m <= l; m++) {
    double sg = (m & 1) ? -1.0 : 1.0;
    q[(l+m)*d + (l+m)] = { sg*is2, 0.0 };
    q[(l+m)*d + (l-m)] = { 0.0, sg*is2 };
  }
  cplx ph;
  int lm = l & 3;
  if (lm == 0) ph = { 1.0, 0.0 };
  else if (lm == 1) ph = { 0.0, -1.0 };
  else if (lm == 2) ph = { -1.0, 0.0 };
  else ph = { 0.0, 1.0 };
  for (int a = 0; a < d*d; a++) q[a] = cmul(ph, q[a]);
}

__global__ void cg_setup(float* __restrict__ CG_out) {
  int p = threadIdx.x;
  if (p >= 15) return;
  const int pl1[15] = {0,0,0,1,1,1,1,1,1,2,2,2,2,2,2};
  const int pl2[15] = {0,1,2,0,1,1,1,2,2,0,1,1,2,2,2};
  const int pl3[15] = {0,1,2,1,0,1,2,1,2,2,1,2,0,1,2};
  const int cgoff[15] = {0,1,10,35,44,53,80,125,170,245,270,315,390,415,490};
  int l1 = pl1[p], l2 = pl2[p], l3 = pl3[p];
  int d1 = 2*l1+1, d2 = 2*l2+1, d3 = 2*l3+1;
  double C[125];
  su2_cg(l1, l2, l3, C);
  cplx q1[25], q2[25], q3[25];
  qmat(l1, q1); qmat(l2, q2); qmat(l3, q3);
  double Tre[125], Tim[125];
  for (int j = 0; j < d1; j++)
    for (int lq = 0; lq < d2; lq++)
      for (int n = 0; n < d3; n++) {
        cplx acc = { 0.0, 0.0 };
        for (int i2 = 0; i2 < d1; i2++)
          for (int k2 = 0; k2 < d2; k2++) {
            cplx q12 = cmul(q1[i2*d1 + j], q2[k2*d2 + lq]);
            for (int m2 = 0; m2 < d3; m2++) {
              double cc = C[(i2*d2 + k2)*d3 + m2];
              if (cc == 0.0) continue;
              cplx q3c = { q3[m2*d3 + n].re, -q3[m2*d3 + n].im };
              cplx t = cmul(q12, q3c);
              acc.re += t.re * cc;
              acc.im += t.im * cc;
            }
          }
        Tre[(j*d2 + lq)*d3 + n] = acc.re;
        Tim[(j*d2 + lq)*d3 + n] = acc.im;
      }
  int sz = d1*d2*d3;
  double nr = 0.0;
  for (int a = 0; a < sz; a++) nr += Tre[a]*Tre[a];
  nr = sqrt(nr);
  const double* src = Tre;
  if (nr < 1e-6) {
    double ni = 0.0;
    for (int a = 0; a < sz; a++) ni += Tim[a]*Tim[a];
    nr = sqrt(ni);
    src = Tim;
  }
  float* o = CG_out + cgoff[p];
  for (int a = 0; a < sz; a++) o[a] = (float)(src[a] / nr);
}

// ================= specialized tensor-product path =================
// Stage 1 (WMMA): R[t][v*32+w] = sum_u y_l1[t,u,i] * Wp[u][v][w]
//   M=16 tokens, K=32 (u), N=1024 (v*32+w); 64 N-tiles, 16 per wave, 2 in flight.
// Stage 2 (VALU): z[t,w,k] += sum_v R[t,v,w] * g[t,v,k],
//   g[t,v,k] = coeff * sum_j CG[i,j,k] * y_l2[t,v,j].
template<int L1, int L2, int L3, int P, int CGOFF>
__device__ __forceinline__ void tp_path(
    const float* __restrict__ Wtp,
    const float* sY, float* sR, float* sG, const float* sCG,
    float (&zacc)[4][9],
    int tid, int wave, int lane, int laneM, int hi)
{
  constexpr int d1 = 2*L1+1, d2 = 2*L2+1, d3 = 2*L3+1;
  constexpr int off1 = (L1 == 0) ? 0 : ((L1 == 1) ? 32 : 128);
  constexpr int off2 = (L2 == 0) ? 0 : ((L2 == 1) ? 32 : 128);
  constexpr int zb   = (L3 == 0) ? 0 : ((L3 == 1) ? 1 : 4);
  constexpr int NTO  = (L3 == 0) ? 3 : 6;
  const float coeff = sqrtf((float)(2*L3+1) / (1024.0f * (float)NTO));
  const float* __restrict__ Wp = Wtp + (long)P * 32768;

  // warm the next path's weight slab (128 KB / 128 threads = 1 KB each)
  if (P + 1 < 15)
    __builtin_prefetch((const char*)(Wtp + (long)(P+1) * 32768) + tid * 1024, 0, 1);

  // per-lane B base: row pair selected by lane half (hi), column by laneM
  const float* __restrict__ bbase = Wp + laneM + hi*2048;

  #pragma unroll 1
  for (int i = 0; i < d1; i++) {
    // ---- A fragments (shared by all 16 N-tiles of this wave) ----
    const float* arow = sY + laneM*YSTR + off1 + i;
    float ax[8], ay[8];
    #pragma unroll
    for (int ks = 0; ks < 8; ks++) {
      const int u0 = ks*4 + hi*2;
      ax[ks] = arow[u0*d1];
      ay[ks] = arow[(u0+1)*d1];
    }
    // ---- stage 1: two N-tiles in flight per iteration ----
    #pragma unroll 1
    for (int nt = wave; nt < 64; nt += 8) {
      const float* bp0 = bbase + nt*16;
      const float* bp1 = bbase + (nt+4)*16;
      v8f acc0 = {}, acc1 = {};
      #pragma unroll
      for (int ks = 0; ks < 8; ks++) {
        v2f a;  a.x = ax[ks];          a.y = ay[ks];
        v2f b0; b0.x = bp0[ks*4096];   b0.y = bp0[ks*4096 + 1024];
        v2f b1; b1.x = bp1[ks*4096];   b1.y = bp1[ks*4096 + 1024];
        acc0 = __builtin_amdgcn_wmma_f32_16x16x4_f32(false, a, false, b0,
                                                     (short)0, acc0, false, false);
        acc1 = __builtin_amdgcn_wmma_f32_16x16x4_f32(false, a, false, b1,
                                                     (short)0, acc1, false, false);
      }
      float* r0 = sR + (hi*8)*RSTR + nt*16 + laneM;
      float* r1 = r0 + 4*16;
      #pragma unroll
      for (int g = 0; g < 8; g++) { r0[g*RSTR] = acc0[g]; r1[g*RSTR] = acc1[g]; }
    }
    // ---- g[t][v][k] = coeff * sum_j CG[i,j,k] * y_l2[t,v,j] ----
    {
      const float* cgi = sCG + CGOFF + i*(d2*d3);
      #pragma unroll 1
      for (int r = 0; r < 4*d3; r++) {
        const int idx = r*NTHR + tid;            // 16*32*d3 total
        const int k = idx % d3;
        const int v = (idx / d3) & 31;
        const int t = idx / (32*d3);
        const float* yrow = sY + t*YSTR + off2 + v*d2;
        float gacc = 0.0f;
        #pragma unroll
        for (int j = 0; j < d2; j++)
          gacc += cgi[j*d3 + k] * yrow[j];
        sG[(t*32 + v)*5 + k] = gacc * coeff;
      }
    }
    __syncthreads();
    // ---- stage 2 ----
    #pragma unroll
    for (int j4 = 0; j4 < 4; j4++) {
      const int t = wave + 4*j4;
      const float* rrow = sR + t*RSTR + lane;
      const float* grow = sG + t*32*5;
      float zk[d3] = {};
      #pragma unroll 4
      for (int v = 0; v < 32; v++) {
        const float r = rrow[v*32];
        #pragma unroll
        for (int k = 0; k < d3; k++) zk[k] += r * grow[v*5 + k];
      }
      #pragma unroll
      for (int k = 0; k < d3; k++) zacc[j4][zb + k] += zk[k];
    }
    __syncthreads();
  }
}

// ================= main fused kernel =================
__global__ __launch_bounds__(NTHR) void shffn_main(
    const float* __restrict__ x,  const float* __restrict__ W1,
    const float* __restrict__ Wtp, const float* __restrict__ W2,
    const float* __restrict__ CG,  float* __restrict__ out, int ntok)
{
  __shared__ float sXZ[TILE*YSTR];      // x tile; later reused as z staging
  __shared__ float sY [TILE*YSTR];      // lin1 output
  __shared__ float sR [TILE*RSTR];      // stage-1 WMMA result (padded rows)
  __shared__ float sG [TILE*32*5];      // g[t][v][k]
  __shared__ float sW [6144];           // W1 (0..3071), W2 (3072..6143)
  __shared__ float sCG[615];

  const int tid   = threadIdx.x;
  const int wave  = tid >> 5;
  const int lane  = tid & 31;
  const int laneM = lane & 15;
  const int hi    = lane >> 4;
  const long tok0 = (long)blockIdx.x * TILE;
  const float s_lin = 0.17677669529663687f;   // 1/sqrt(32)

  // ---- stage weights / CG / x tile into LDS ----
  for (int idx = tid; idx < 6144; idx += NTHR)
    sW[idx] = (idx < 3072) ? W1[idx] : W2[idx - 3072];
  for (int idx = tid; idx < 615; idx += NTHR) sCG[idx] = CG[idx];
  for (int idx = tid; idx < TILE*DIM_IR; idx += NTHR) {
    int t = idx / DIM_IR, c = idx % DIM_IR;
    long tg = tok0 + t;
    sXZ[t*YSTR + c] = (tg < (long)ntok) ? x[tg * DIM_IR + c] : 0.0f;
  }
  __syncthreads();

  // ---- lin1 (VALU; tiny), per-l specialized ----
  {
    const float* w0 = &sW[0];
    for (int idx = tid; idx < TILE*32; idx += NTHR) {        // l=0
      int t = idx >> 5, wch = idx & 31;
      float acc = 0.0f;
      #pragma unroll 8
      for (int u = 0; u < 32; u++) acc += sXZ[t*YSTR + u] * w0[u*32 + wch];
      sY[t*YSTR + wch] = acc * s_lin;
    }
    const float* w1 = &sW[1024];
    for (int idx = tid; idx < TILE*96; idx += NTHR) {        // l=1
      int t = idx / 96, r = idx % 96, wch = r / 3, i = r % 3;
      float acc = 0.0f;
      #pragma unroll 8
      for (int u = 0; u < 32; u++) acc += sXZ[t*YSTR + 32 + u*3 + i] * w1[u*32 + wch];
      sY[t*YSTR + 32 + wch*3 + i] = acc * s_lin;
    }
    const float* w2 = &sW[2048];
    for (int idx = tid; idx < TILE*160; idx += NTHR) {       // l=2
      int t = idx / 160, r = idx % 160, wch = r / 5, i = r % 5;
      float acc = 0.0f;
      #pragma unroll 8
      for (int u = 0; u < 32; u++) acc += sXZ[t*YSTR + 128 + u*5 + i] * w2[u*32 + wch];
      sY[t*YSTR + 128 + wch*5 + i] = acc * s_lin;
    }
  }
  __syncthreads();

  // ---- tensor product: 15 compile-time-specialized CG paths ----
  float zacc[4][9];
  #pragma unroll
  for (int a = 0; a < 4; a++)
    #pragma unroll
    for (int b = 0; b < 9; b++) zacc[a][b] = 0.0f;

  tp_path<0,0,0, 0,   0>(Wtp, sY, sR, sG, sCG, zacc, tid, wave, lane, laneM, hi);
  tp_path<0,1,1, 1,   1>(Wtp, sY, sR, sG, sCG, zacc, tid, wave, lane, laneM, hi);
  tp_path<0,2,2, 2,  10>(Wtp, sY, sR, sG, sCG, zacc, tid, wave, lane, laneM, hi);
  tp_path<1,0,1, 3,  35>(Wtp, sY, sR, sG, sCG, zacc, tid, wave, lane, laneM, hi);
  tp_path<1,1,0, 4,  44>(Wtp, sY, sR, sG, sCG, zacc, tid, wave, lane, laneM, hi);
  tp_path<1,1,1, 5,  53>(Wtp, sY, sR, sG, sCG, zacc, tid, wave, lane, laneM, hi);
  tp_path<1,1,2, 6,  80>(Wtp, sY, sR, sG, sCG, zacc, tid, wave, lane, laneM, hi);
  tp_path<1,2,1, 7, 125>(Wtp, sY, sR, sG, sCG, zacc, tid, wave, lane, laneM, hi);
  tp_path<1,2,2, 8, 170>(Wtp, sY, sR, sG, sCG, zacc, tid, wave, lane, laneM, hi);
  tp_path<2,0,2, 9, 245>(Wtp, sY, sR, sG, sCG, zacc, tid, wave, lane, laneM, hi);
  tp_path<2,1,1,10, 270>(Wtp, sY, sR, sG, sCG, zacc, tid, wave, lane, laneM, hi);
  tp_path<2,1,2,11, 315>(Wtp, sY, sR, sG, sCG, zacc, tid, wave, lane, laneM, hi);
  tp_path<2,2,0,12, 390>(Wtp, sY, sR, sG, sCG, zacc, tid, wave, lane, laneM, hi);
  tp_path<2,2,1,13, 415>(Wtp, sY, sR, sG, sCG, zacc, tid, wave, lane, laneM, hi);
  tp_path<2,2,2,14, 490>(Wtp, sY, sR, sG, sCG, zacc, tid, wave, lane, laneM, hi);

  // ---- dump z into LDS (reuse sXZ), e3nn layout [t][off_l + u*(2l+1)+k] ----
  #pragma unroll
  for (int j4 = 0; j4 < 4; j4++) {
    const int t = wave + 4*j4;
    const int u = lane;
    sXZ[t*YSTR + u] = zacc[j4][0];
    #pragma unroll
    for (int k = 0; k < 3; k++) sXZ[t*YSTR + 32  + u*3 + k] = zacc[j4][1 + k];
    #pragma unroll
    for (int k = 0; k < 5; k++) sXZ[t*YSTR + 128 + u*5 + k] = zacc[j4][4 + k];
  }
  __syncthreads();

  // ---- lin2 (VALU; tiny) -> global, per-l specialized ----
  {
    const float* w0 = &sW[3072];
    for (int idx = tid; idx < TILE*32; idx += NTHR) {        // l=0
      int t = idx >> 5, wch = idx & 31;
      long tg = tok0 + t;
      if (tg >= (long)ntok) continue;
      float acc = 0.0f;
      #pragma unroll 8
      for (int u = 0; u < 32; u++) acc += sXZ[t*YSTR + u] * w0[u*32 + wch];
      out[tg*DIM_IR + wch] = acc * s_lin;
    }
    const float* w1 = &sW[3072 + 1024];
    for (int idx = tid; idx < TILE*96; idx += NTHR) {        // l=1
      int t = idx / 96, r = idx % 96, wch = r / 3, i = r % 3;
      long tg = tok0 + t;
      if (tg >= (long)ntok) continue;
      float acc = 0.0f;
      #pragma unroll 8
      for (int u = 0; u < 32; u++) acc += sXZ[t*YSTR + 32 + u*3 + i] * w1[u*32 + wch];
      out[tg*DIM_IR + 32 + wch*3 + i] = acc * s_lin;
    }
    const float* w2 = &sW[3072 + 2048];
    for (int idx = tid; idx < TILE*160; idx += NTHR) {       // l=2
      int t = idx / 160, r = idx % 160, wch = r / 5, i = r % 5;
      long tg = tok0 + t;
      if (tg >= (long)ntok) continue;
      float acc = 0.0f;
      #pragma unroll 8
      for (int u = 0; u < 32; u++) acc += sXZ[t*YSTR + 128 + u*5 + i] * w2[u*32 + wch];
      out[tg*DIM_IR + 128 + wch*5 + i] = acc * s_lin;
    }
  }
}

// ================= launcher =================
extern "C" void kernel_launch(void* const* d_in, const int* in_sizes, int n_in,
                              void* d_out, int out_size, void* d_ws, size_t ws_size,
                              hipStream_t stream) {
  const float* x   = (const float*)d_in[0];
  const float* W1  = (const float*)d_in[1];
  const float* Wtp = (const float*)d_in[2];
  const float* W2  = (const float*)d_in[3];
  float* out = (float*)d_out;
  float* cg  = (float*)d_ws;               // 615 floats of CG coefficients
  int ntok = in_sizes[0] / DIM_IR;

  cg_setup<<<1, 32, 0, stream>>>(cg);
  int nblk = (ntok + TILE - 1) / TILE;
  shffn_main<<<nblk, NTHR, 0, stream>>>(x, W1, Wtp, W2, cg, out, ntok);
}